// SDE_GPSSM_57440892617109
// MI455X (gfx1250) — compile-verified
//
#include <hip/hip_runtime.h>
#include <math.h>

// ---------------------------------------------------------------------------
// SDE-GPSSM pipeline for MI455X (gfx1250, wave32, WGP, 320KB LDS).
//   N=8192 timesteps, M=128 obs dim, D=64 state dim, HID=32.
// d_out layout (flat, return order):
//   smooth_means [8192*64]      @ 0
//   smooth_covs  [8192*64*64]   @ 524288
//   pseudo_y     [8192*128]     @ 34078720
//   pseudo_var   [8192*128]     @ 35127296
//   log_lik      [1]            @ 36175872
// d_ws: Fs = 8192 * 64*64 floats = 128 MiB (assumed available).
// CDNA5 paths used: v_wmma_f32_16x16x4_f32 for every dense matmul,
// tensor_load_to_lds / tensor_store_from_lds (TDM) for tile DMA with
// s_wait_tensorcnt, double-buffered F in both sequential scans.
// ---------------------------------------------------------------------------

typedef float v2f __attribute__((ext_vector_type(2)));
typedef float v8f __attribute__((ext_vector_type(8)));
typedef unsigned int v4u __attribute__((ext_vector_type(4)));
typedef int v4i __attribute__((ext_vector_type(4)));
typedef int v8i __attribute__((ext_vector_type(8)));

#define NSTEP 8192
#define MDIM  128
#define DDIM  64
#define HID   32
#define LOG2PI 1.8378770664093453f

// fp32 WMMA 16x16x4 (CDNA5): D = A(16x4) * B(4x16) + C(16x16)
__device__ __forceinline__ v8f wmma_f32(v2f a, v2f b, v8f c) {
  return __builtin_amdgcn_wmma_f32_16x16x4_f32(
      /*neg_a=*/false, a, /*neg_b=*/false, b,
      /*c_mod=*/(short)0, c, /*reuse_a=*/false, /*reuse_b=*/false);
}

// ---------------------------------------------------------------------------
// Tensor Data Mover: 1-D contiguous f32 tile between global and LDS.
// D# group0: count=1, lds_addr, global_addr, type=2.
// D# group1: data_size=4B, tensor_dim0 = tile_dim0 = nelem (1-D tile),
//            tensor_dim1 = 1, stride0 = nelem, no padding / multicast.
// ---------------------------------------------------------------------------
__device__ __forceinline__ void tdm_desc_1d(unsigned lds_off, const void* gptr,
                                            unsigned nelem, v4u& g0, v8i& g1) {
  const unsigned long long ga = (unsigned long long)gptr;
  g0[0] = 1u;                                               // count=1 (user)
  g0[1] = lds_off;                                          // lds_addr bytes
  g0[2] = (unsigned)(ga & 0xffffffffu);                     // global_addr lo
  g0[3] = (unsigned)((ga >> 32) & 0x01ffffffu) | (2u << 30);// addr hi | type=2
  g1[0] = (int)(2u << 16);                                  // data_size = 4B
  g1[1] = (int)((nelem & 0xffffu) << 16);                   // tensor_dim0 lo16
  g1[2] = (int)(((nelem >> 16) & 0xffffu) | (1u << 16));    // dim0 hi, dim1=1
  g1[3] = (int)((nelem & 0xffffu) << 16);                   // tile_dim0
  g1[4] = 0;                                                // tile_dim1/2 = 0
  g1[5] = (int)nelem;                                       // stride0 lo32
  g1[6] = 0;
  g1[7] = 0;
}

__device__ __forceinline__ void tdm_load_1d(unsigned lds_off, const void* gptr,
                                            unsigned nelem) {
  v4u g0; v8i g1;
  tdm_desc_1d(lds_off, gptr, nelem, g0, g1);
  const v4i z = {0, 0, 0, 0};
#if defined(__clang_major__) && __clang_major__ >= 23
  const v8i z8 = {0, 0, 0, 0, 0, 0, 0, 0};
  __builtin_amdgcn_tensor_load_to_lds(g0, g1, z, z, z8, 0);
#else
  __builtin_amdgcn_tensor_load_to_lds(g0, g1, z, z, 0);
#endif
}

__device__ __forceinline__ void tdm_store_1d(unsigned lds_off, void* gptr,
                                             unsigned nelem) {
  v4u g0; v8i g1;
  tdm_desc_1d(lds_off, gptr, nelem, g0, g1);
  const v4i z = {0, 0, 0, 0};
#if defined(__clang_major__) && __clang_major__ >= 23
  const v8i z8 = {0, 0, 0, 0, 0, 0, 0, 0};
  __builtin_amdgcn_tensor_store_from_lds(g0, g1, z, z, z8, 0);
#else
  __builtin_amdgcn_tensor_store_from_lds(g0, g1, z, z, 0);
#endif
}

// ---------------------------------------------------------------------------
// C(MxN) = op(A)(MxK) * op(B)(KxN), row-major, via fp32 WMMA.
// TA: A stored KxM (use A^T). TB: B stored NxK (use B^T).
// K is processed in 64-deep chunks: all 16 A/B fragments of a chunk are
// preloaded into VGPRs, then 16 wmmas issue back-to-back (memory-level
// parallelism instead of load->wait->wmma per step).
// Tiles distributed over the block's waves; caller handles barriers.
// ---------------------------------------------------------------------------
template <int M, int N, int K, bool TA, bool TB>
__device__ void mm(const float* A, int lda, const float* B, int ldb,
                   float* C, int ldc) {
  const int lane = threadIdx.x & 31;
  const int wave = threadIdx.x >> 5;
  const int nw   = blockDim.x >> 5;
  const int r = lane & 15;
  const int h = lane >> 4;
  constexpr int TM = M / 16, TN = N / 16;
  constexpr int CH = (K < 64) ? K : 64;
  constexpr int CF = CH / 4;
  for (int t = wave; t < TM * TN; t += nw) {
    const int tm = t / TN, tn = t % TN;
    const int row = tm * 16 + r;
    const int col = tn * 16 + r;
    v8f acc = {};
    for (int k0 = 0; k0 < K; k0 += CH) {
      v2f af[CF], bf[CF];
#pragma unroll
      for (int i = 0; i < CF; ++i) {
        const int ka = k0 + i * 4 + 2 * h;
        if (TA) { af[i].x = A[ka * lda + row];  af[i].y = A[(ka + 1) * lda + row]; }
        else    { af[i].x = A[row * lda + ka];  af[i].y = A[row * lda + ka + 1];   }
        if (TB) { bf[i].x = B[col * ldb + ka];  bf[i].y = B[col * ldb + ka + 1];   }
        else    { bf[i].x = B[ka * ldb + col];  bf[i].y = B[(ka + 1) * ldb + col]; }
      }
#pragma unroll
      for (int i = 0; i < CF; ++i) acc = wmma_f32(af[i], bf[i], acc);
    }
#pragma unroll
    for (int q = 0; q < 8; ++q)
      C[(tm * 16 + q + 8 * h) * ldc + col] = acc[q];
  }
}

// In-place lower Cholesky of NNxNN SPD matrix in LDS.
template <int NN>
__device__ void cholesky_lds(float* S) {
  const int tid = threadIdx.x, nt = blockDim.x;
  for (int j = 0; j < NN; ++j) {
    if (tid == 0) S[j * NN + j] = sqrtf(S[j * NN + j]);
    __syncthreads();
    const float d = S[j * NN + j];
    for (int i = j + 1 + tid; i < NN; i += nt) S[i * NN + j] /= d;
    __syncthreads();
    const int rr = NN - 1 - j;
    for (int idx = tid; idx < rr * rr; idx += nt) {
      const int i = j + 1 + idx / rr, k2 = j + 1 + idx % rr;
      if (k2 <= i) S[i * NN + k2] -= S[i * NN + j] * S[k2 * NN + j];
    }
    __syncthreads();
  }
}

// Solve (L L^T) X = Bm for NNxNR RHS in LDS, in place (column-sweep).
template <int NN, int NR>
__device__ void cho_solve_lds(const float* L, float* Bm) {
  const int tid = threadIdx.x, nt = blockDim.x;
  for (int j = 0; j < NN; ++j) {            // forward: L y = B
    if (tid < NR) Bm[j * NR + tid] /= L[j * NN + j];
    __syncthreads();
    const int rows = NN - 1 - j;
    for (int idx = tid; idx < rows * NR; idx += nt) {
      const int i = j + 1 + idx / NR, c = idx % NR;
      Bm[i * NR + c] -= L[i * NN + j] * Bm[j * NR + c];
    }
    __syncthreads();
  }
  for (int j = NN - 1; j >= 0; --j) {       // backward: L^T x = y
    if (tid < NR) Bm[j * NR + tid] /= L[j * NN + j];
    __syncthreads();
    for (int idx = tid; idx < j * NR; idx += nt) {
      const int i = idx / NR, c = idx % NR;
      Bm[i * NR + c] -= L[j * NN + i] * Bm[j * NR + c];
    }
    __syncthreads();
  }
}

// ---------------------------------------------------------------------------
// Kernel 1: MLP encoder.  64 rows of Y per block; Y tile DMA'd by the TDM.
// ---------------------------------------------------------------------------
__global__ __launch_bounds__(256) void mlp_kernel(
    const float* __restrict__ Y, const float* __restrict__ W1,
    const float* __restrict__ b1, const float* __restrict__ W2,
    const float* __restrict__ b2, const float* __restrict__ W3,
    const float* __restrict__ b3, float* __restrict__ py,
    float* __restrict__ pv) {
  extern __shared__ float smem[];
  float* Yt = smem;                 // 64 x 128
  float* Hh = Yt + 64 * MDIM;       // 64 x 32
  float* Ot = Hh + 64 * HID;        // 64 x 128
  const int tid = threadIdx.x, nt = blockDim.x;
  const int wid = tid >> 5;
  const int row0 = blockIdx.x * 64;

  if (wid == 0) {
    tdm_load_1d((unsigned)((char*)Yt - (char*)smem),
                Y + (size_t)row0 * MDIM, 64 * MDIM);
    __builtin_amdgcn_s_wait_tensorcnt(0);
  }
  __syncthreads();
  mm<64, HID, MDIM, false, true>(Yt, MDIM, W1, MDIM, Hh, HID);
  __syncthreads();
  for (int i = tid; i < 64 * HID; i += nt) {
    const float v = Hh[i] + b1[i & (HID - 1)];
    Hh[i] = v > 0.0f ? v : 0.0f;
  }
  __syncthreads();
  mm<64, MDIM, HID, false, true>(Hh, HID, W2, HID, Ot, MDIM);
  __syncthreads();
  for (int i = tid; i < 64 * MDIM; i += nt)
    py[(size_t)row0 * MDIM + i] = Ot[i] + b2[i & (MDIM - 1)];
  __syncthreads();
  mm<64, MDIM, HID, false, true>(Hh, HID, W3, HID, Ot, MDIM);
  __syncthreads();
  for (int i = tid; i < 64 * MDIM; i += nt) {
    const float x = Ot[i] + b3[i & (MDIM - 1)];
    pv[(size_t)row0 * MDIM + i] = (x > 20.0f) ? x : log1pf(expf(x));
  }
}

// ---------------------------------------------------------------------------
// Kernel 2: batched expm.  One block per timestep: scale by 1/16, Taylor-8,
// 4 squarings, all 64x64 matmuls via fp32 WMMA; result DMA'd out by the TDM.
// ---------------------------------------------------------------------------
__global__ __launch_bounds__(256) void expm_kernel(
    const float* __restrict__ A, const float* __restrict__ dt,
    float* __restrict__ Fs) {
  extern __shared__ float smem[];
  float* X = smem;          // scaled matrix
  float* E = X + 4096;      // exp accumulator
  float* T = E + 4096;      // current Taylor term
  float* W = T + 4096;      // matmul scratch
  const int tid = threadIdx.x, nt = blockDim.x;
  const int wid = tid >> 5;
  const int t = blockIdx.x;
  const float s = dt[t] * (1.0f / 16.0f);

  for (int i = tid; i < 4096; i += nt) {
    X[i] = A[i] * s;
    const float id = ((i >> 6) == (i & 63)) ? 1.0f : 0.0f;
    E[i] = id;
    T[i] = id;
  }
  __syncthreads();
  for (int k = 1; k <= 8; ++k) {
    mm<DDIM, DDIM, DDIM, false, false>(T, DDIM, X, DDIM, W, DDIM);
    __syncthreads();
    const float inv = 1.0f / (float)k;
    for (int i = tid; i < 4096; i += nt) {
      const float v = W[i] * inv;
      T[i] = v;
      E[i] += v;
    }
    __syncthreads();
  }
  for (int q = 0; q < 4; ++q) {
    mm<DDIM, DDIM, DDIM, false, false>(E, DDIM, E, DDIM, W, DDIM);
    __syncthreads();
    for (int i = tid; i < 4096; i += nt) E[i] = W[i];
    __syncthreads();
  }
  // async DMA of the 16KB result; wave 0 holds the workgroup until done
  if (wid == 0) {
    tdm_store_1d((unsigned)((char*)E - (char*)smem),
                 Fs + (size_t)t * 4096, 4096);
    __builtin_amdgcn_s_wait_tensorcnt(0);
  }
}

// ---------------------------------------------------------------------------
// Kernel 3: sequential Kalman filter scan.  Single 1024-thread workgroup
// (32 wave32s), all per-step state LDS-resident (~212KB: needs CDNA5's
// 320KB WGP LDS).  F[t] is double-buffered via TDM: wave 0 issues the DMA
// for step t+1 right after step t's s_wait_tensorcnt retires, so HBM latency
// hides behind the 128x128 Cholesky chain.  Filtered m/P land in-place in
// the smooth_means/smooth_covs output regions.
// ---------------------------------------------------------------------------
__global__ __launch_bounds__(1024) void kf_forward_kernel(
    const float* __restrict__ Hg, const float* __restrict__ Bg,
    const float* __restrict__ m0, const float* __restrict__ dt,
    const float* __restrict__ Fs, const float* __restrict__ py,
    const float* __restrict__ pv, float* __restrict__ sm_out,
    float* __restrict__ sc_out, float* __restrict__ ll_out) {
  extern __shared__ float smem[];
  float* F0  = smem;            // F double buffer (TDM target)
  float* F1  = F0  + 4096;
  float* P   = F1  + 4096;      // carry covariance
  float* Pn  = P   + 4096;      // predicted covariance
  float* T64 = Pn  + 4096;      // F @ P
  float* Qml = T64 + 4096;      // Qm = B B^T
  float* G   = Qml + 4096;      // K H  ->  I - K H
  float* HP  = G   + 4096;      // 128x64: H Pn, then X = S^{-1}(H Pn)
  float* S   = HP  + 8192;      // 128x128: innovation cov -> chol L
  float* vec = S   + 16384;
  float* m   = vec;             // 64   carry mean
  float* mp  = vec + 64;        // 64   predicted mean
  float* yv  = vec + 128;       // 128
  float* rv  = vec + 256;       // 128
  float* inn = vec + 384;       // 128
  float* res = vec + 512;       // 128
  float* zv  = vec + 640;       // 128

  const int tid = threadIdx.x, nt = blockDim.x;
  const int wid = tid >> 5;
  const unsigned f0_off = (unsigned)((char*)F0 - (char*)smem);
  const unsigned f1_off = (unsigned)((char*)F1 - (char*)smem);

  mm<DDIM, DDIM, DDIM, false, true>(Bg, DDIM, Bg, DDIM, Qml, DDIM);  // Qm
  for (int i = tid; i < 4096; i += nt)
    P[i] = ((i >> 6) == (i & 63)) ? 0.1f : 0.0f;
  if (tid < DDIM) m[tid] = m0[tid];
  if (wid == 0) tdm_load_1d(f0_off, Fs, 4096);          // F[0] -> buf0
  __syncthreads();

  float llsum = 0.0f;
  for (int t = 0; t < NSTEP; ++t) {
    const float dts = dt[t];
    if (wid == 0) {
      __builtin_amdgcn_s_wait_tensorcnt(0);             // F[t] resident
      if (t + 1 < NSTEP)                                // prefetch F[t+1]
        tdm_load_1d(((t + 1) & 1) ? f1_off : f0_off,
                    Fs + (size_t)(t + 1) * 4096, 4096);
    }
    if (tid < MDIM) {
      yv[tid] = py[(size_t)t * MDIM + tid];
      rv[tid] = pv[(size_t)t * MDIM + tid];
    }
    __syncthreads();
    const float* Fc = (t & 1) ? F1 : F0;

    if (tid < DDIM) {                       // mp = F m
      float acc = 0.0f;
      for (int j = 0; j < DDIM; ++j) acc += Fc[tid * DDIM + j] * m[j];
      mp[tid] = acc;
    }
    __syncthreads();
    mm<DDIM, DDIM, DDIM, false, false>(Fc, DDIM, P, DDIM, T64, DDIM);
    __syncthreads();
    mm<DDIM, DDIM, DDIM, false, true>(T64, DDIM, Fc, DDIM, Pn, DDIM);
    __syncthreads();
    for (int i = tid; i < 4096; i += nt) Pn[i] += Qml[i] * dts;
    __syncthreads();
    mm<MDIM, DDIM, DDIM, false, false>(Hg, DDIM, Pn, DDIM, HP, DDIM);
    __syncthreads();
    mm<MDIM, MDIM, DDIM, false, true>(HP, DDIM, Hg, DDIM, S, MDIM);
    __syncthreads();
    if (tid < MDIM) S[tid * MDIM + tid] += rv[tid];
    __syncthreads();
    cholesky_lds<MDIM>(S);
    cho_solve_lds<MDIM, DDIM>(S, HP);       // HP := X, K = X^T
    if (tid < MDIM) {                       // innov = y - H mp
      float acc = 0.0f;
      for (int j = 0; j < DDIM; ++j) acc += Hg[tid * DDIM + j] * mp[j];
      inn[tid] = yv[tid] - acc;
    }
    __syncthreads();
    if (tid < DDIM) {                       // m = mp + K innov
      float acc = 0.0f;
      for (int j = 0; j < MDIM; ++j) acc += HP[j * DDIM + tid] * inn[j];
      m[tid] = mp[tid] + acc;
    }
    __syncthreads();
    mm<DDIM, DDIM, MDIM, true, false>(HP, DDIM, Hg, DDIM, G, DDIM);  // K H
    __syncthreads();
    for (int i = tid; i < 4096; i += nt)
      G[i] = (((i >> 6) == (i & 63)) ? 1.0f : 0.0f) - G[i];
    __syncthreads();
    mm<DDIM, DDIM, DDIM, false, false>(G, DDIM, Pn, DDIM, P, DDIM);
    __syncthreads();
    if (tid < MDIM) {                       // resid = y - H m
      float acc = 0.0f;
      for (int j = 0; j < DDIM; ++j) acc += Hg[tid * DDIM + j] * m[j];
      const float rr = yv[tid] - acc;
      res[tid] = rr;
      zv[tid] = rr;
    }
    __syncthreads();
    cho_solve_lds<MDIM, 1>(S, zv);
    if (tid == 0) {
      float quad = 0.0f, logdet = 0.0f;
      for (int j = 0; j < MDIM; ++j) {
        quad += res[j] * zv[j];
        logdet += __logf(S[j * MDIM + j]);
      }
      llsum += -0.5f * ((float)MDIM * LOG2PI + 2.0f * logdet + quad);
    }
    if (tid < DDIM) sm_out[(size_t)t * DDIM + tid] = m[tid];
    for (int i = tid; i < 4096; i += nt) sc_out[(size_t)t * 4096 + i] = P[i];
    __syncthreads();
  }
  if (tid == 0) ll_out[0] = llsum;
}

// ---------------------------------------------------------------------------
// Kernel 4: backward RTS smoother, in-place on the output arrays.
// Fs_s[t] == Fs[t+1] (dt_s = dt[1:]); F double-buffered via TDM as above.
// ---------------------------------------------------------------------------
__global__ __launch_bounds__(1024) void rts_kernel(
    const float* __restrict__ Bg, const float* __restrict__ dt,
    const float* __restrict__ Fs, float* __restrict__ sm_out,
    float* __restrict__ sc_out) {
  extern __shared__ float smem[];
  float* F0  = smem;
  float* F1  = F0  + 4096;
  float* Pf  = F1  + 4096;
  float* T64 = Pf  + 4096;   // F Pf, then X = Ppred^{-1}(F Pf), G = X^T
  float* Pp  = T64 + 4096;   // P_pred -> chol L -> G Dm G^T
  float* Dm  = Pp  + 4096;   // Psn - P_pred
  float* T2  = Dm  + 4096;   // G Dm
  float* Psn = T2  + 4096;   // carry smoothed covariance
  float* Qml = Psn + 4096;
  float* vec = Qml + 4096;
  float* msn = vec;          // 64 carry smoothed mean
  float* mf  = vec + 64;
  float* vv  = vec + 128;
  float* wv  = vec + 192;
  float* msv = vec + 256;

  const int tid = threadIdx.x, nt = blockDim.x;
  const int wid = tid >> 5;
  const unsigned f0_off = (unsigned)((char*)F0 - (char*)smem);
  const unsigned f1_off = (unsigned)((char*)F1 - (char*)smem);

  mm<DDIM, DDIM, DDIM, false, true>(Bg, DDIM, Bg, DDIM, Qml, DDIM);
  for (int i = tid; i < 4096; i += nt)
    Psn[i] = sc_out[(size_t)(NSTEP - 1) * 4096 + i];
  if (tid < DDIM) msn[tid] = sm_out[(size_t)(NSTEP - 1) * DDIM + tid];
  // first iteration (t = NSTEP-2, even) uses Fs[NSTEP-1] from buf0
  if (wid == 0)
    tdm_load_1d(f0_off, Fs + (size_t)(NSTEP - 1) * 4096, 4096);
  __syncthreads();

  for (int t = NSTEP - 2; t >= 0; --t) {
    const float dts = dt[t + 1];
    if (wid == 0) {
      __builtin_amdgcn_s_wait_tensorcnt(0);             // Fs[t+1] resident
      if (t > 0)                                        // prefetch Fs[t]
        tdm_load_1d(((t - 1) & 1) ? f1_off : f0_off,
                    Fs + (size_t)t * 4096, 4096);
    }
    for (int i = tid; i < 4096; i += nt)
      Pf[i] = sc_out[(size_t)t * 4096 + i];
    if (tid < DDIM) mf[tid] = sm_out[(size_t)t * DDIM + tid];
    __syncthreads();
    const float* Fc = (t & 1) ? F1 : F0;
    if (t > 0 && tid < 128)
      __builtin_prefetch(&sc_out[(size_t)(t - 1) * 4096 + tid * 32], 0, 1);
    mm<DDIM, DDIM, DDIM, false, false>(Fc, DDIM, Pf, DDIM, T64, DDIM);
    __syncthreads();
    mm<DDIM, DDIM, DDIM, false, true>(T64, DDIM, Fc, DDIM, Pp, DDIM);
    __syncthreads();
    for (int i = tid; i < 4096; i += nt) {
      const float pp = Pp[i] + Qml[i] * dts;
      Pp[i] = pp;
      Dm[i] = Psn[i] - pp;
    }
    __syncthreads();
    cholesky_lds<DDIM>(Pp);
    cho_solve_lds<DDIM, DDIM>(Pp, T64);     // T64 := X, G = X^T
    if (tid < DDIM) {                       // v = F mf
      float acc = 0.0f;
      for (int j = 0; j < DDIM; ++j) acc += Fc[tid * DDIM + j] * mf[j];
      vv[tid] = acc;
    }
    __syncthreads();
    if (tid < DDIM) wv[tid] = msn[tid] - vv[tid];
    __syncthreads();
    if (tid < DDIM) {                       // m_s = mf + G w
      float acc = 0.0f;
      for (int j = 0; j < DDIM; ++j) acc += T64[j * DDIM + tid] * wv[j];
      msv[tid] = mf[tid] + acc;
    }
    __syncthreads();
    mm<DDIM, DDIM, DDIM, true, false>(T64, DDIM, Dm, DDIM, T2, DDIM);   // G Dm
    __syncthreads();
    mm<DDIM, DDIM, DDIM, false, false>(T2, DDIM, T64, DDIM, Pp, DDIM);  // (G Dm) G^T
    __syncthreads();
    for (int i = tid; i < 4096; i += nt) {
      const float ps = Pf[i] + Pp[i];
      Psn[i] = ps;
      sc_out[(size_t)t * 4096 + i] = ps;
    }
    if (tid < DDIM) {
      msn[tid] = msv[tid];
      sm_out[(size_t)t * DDIM + tid] = msv[tid];
    }
    __syncthreads();
  }
}

// ---------------------------------------------------------------------------
extern "C" void kernel_launch(void* const* d_in, const int* in_sizes, int n_in,
                              void* d_out, int out_size, void* d_ws,
                              size_t ws_size, hipStream_t stream) {
  (void)in_sizes; (void)n_in; (void)out_size; (void)ws_size;
  const float* Y  = (const float*)d_in[0];
  const float* dt = (const float*)d_in[1];
  const float* W1 = (const float*)d_in[2];
  const float* b1 = (const float*)d_in[3];
  const float* W2 = (const float*)d_in[4];
  const float* b2 = (const float*)d_in[5];
  const float* W3 = (const float*)d_in[6];
  const float* b3 = (const float*)d_in[7];
  const float* Hg = (const float*)d_in[8];
  const float* A  = (const float*)d_in[9];
  const float* Bg = (const float*)d_in[10];
  const float* m0 = (const float*)d_in[11];

  float* out = (float*)d_out;
  float* smo = out;                 // 8192*64
  float* sco = out + 524288;        // 8192*4096
  float* pyo = out + 34078720;      // 8192*128
  float* pvo = out + 35127296;      // 8192*128
  float* llo = out + 36175872;      // 1
  float* Fs  = (float*)d_ws;        // 8192*4096 floats = 128 MiB

  mlp_kernel<<<NSTEP / 64, 256, (64 * 128 + 64 * 32 + 64 * 128) * 4, stream>>>(
      Y, W1, b1, W2, b2, W3, b3, pyo, pvo);
  expm_kernel<<<NSTEP, 256, 4 * 4096 * 4, stream>>>(A, dt, Fs);
  kf_forward_kernel<<<1, 1024, (7 * 4096 + 8192 + 16384 + 768) * 4, stream>>>(
      Hg, Bg, m0, dt, Fs, pyo, pvo, smo, sco, llo);
  rts_kernel<<<1, 1024, (9 * 4096 + 320) * 4, stream>>>(Bg, dt, Fs, smo, sco);
}